// M2MGNNPro_26439818674288
// MI455X (gfx1250) — compile-verified
//
#include <hip/hip_runtime.h>
#include <hip/hip_bf16.h>

#define IN_F   128
#define HID_F  64
#define HC_F   128
#define OUT_F  40
#define LN_EPS 1e-5f
#define LDS_STRIDE 132   // 128 + 4 pad; 8*132 % 64 == 32 -> no bank clash between lane halves

typedef __attribute__((ext_vector_type(2))) float v2f;
typedef __attribute__((ext_vector_type(8))) float v8f;

__device__ __forceinline__ void wave_sum2(float& a, float& b) {
    #pragma unroll
    for (int off = 16; off > 0; off >>= 1) {
        a += __shfl_xor(a, off, 32);
        b += __shfl_xor(b, off, 32);
    }
}

// ---------------------------------------------------------------------------
// Kernel 1: h0 = LayerNorm(relu(x @ W1^T + b1)); h = h0 @ Wlin^T  (fused)
// block = 256 threads (8 waves), handles 16 node rows.
// ---------------------------------------------------------------------------
__global__ __launch_bounds__(256) void k_gemm1_ln(
    const float* __restrict__ x, const float* __restrict__ W1,
    const float* __restrict__ b1, const float* __restrict__ g0,
    const float* __restrict__ beta0, const float* __restrict__ Wlin,
    float* __restrict__ h0, float* __restrict__ h)
{
    __shared__ float sh[16 * LDS_STRIDE];
    const int w    = threadIdx.x >> 5;
    const int lane = threadIdx.x & 31;
    const int n    = lane & 15;           // A-row m / B-col n / D-col n
    const int kb   = (lane >> 4) * 2;     // k sub-offset within 4-wide k step
    const int row0 = blockIdx.x * 16;

    // --- GEMM1: this wave computes cols [w*16, w*16+16) over K=128 ---
    {
        const int cb = w * 16;
        v8f acc = {};
        const float* xp = x  + (size_t)(row0 + n) * IN_F;  // A[m][*], m == n
        const float* wp = W1 + (size_t)(cb   + n) * IN_F;  // B[k][n] = W1[cb+n][k]
        for (int k = 0; k < IN_F; k += 4) {
            v2f a, b;
            a.x = xp[k + kb]; a.y = xp[k + kb + 1];
            b.x = wp[k + kb]; b.y = wp[k + kb + 1];
            acc = __builtin_amdgcn_wmma_f32_16x16x4_f32(false, a, false, b,
                                                        (short)0, acc, false, false);
        }
        const float bv = b1[cb + n];
        #pragma unroll
        for (int r = 0; r < 8; ++r) {
            const int m = r + 8 * (lane >> 4);
            sh[m * LDS_STRIDE + cb + n] = fmaxf(acc[r] + bv, 0.f);
        }
    }
    __syncthreads();

    // --- LayerNorm: wave w normalizes rows 2w and 2w+1 (4 elems/lane) ---
    #pragma unroll
    for (int i = 0; i < 2; ++i) {
        const int rr = 2 * w + i;
        float4 v = *reinterpret_cast<const float4*>(&sh[rr * LDS_STRIDE + lane * 4]);
        float s = v.x + v.y + v.z + v.w;
        float q = v.x * v.x + v.y * v.y + v.z * v.z + v.w * v.w;
        wave_sum2(s, q);
        const float mu  = s * (1.f / 128.f);
        const float var = q * (1.f / 128.f) - mu * mu;
        const float rs  = rsqrtf(var + LN_EPS);
        const float4 g  = *reinterpret_cast<const float4*>(&g0[lane * 4]);
        const float4 bb = *reinterpret_cast<const float4*>(&beta0[lane * 4]);
        float4 o;
        o.x = (v.x - mu) * rs * g.x + bb.x;
        o.y = (v.y - mu) * rs * g.y + bb.y;
        o.z = (v.z - mu) * rs * g.z + bb.z;
        o.w = (v.w - mu) * rs * g.w + bb.w;
        *reinterpret_cast<float4*>(&sh[rr * LDS_STRIDE + lane * 4]) = o;
        *reinterpret_cast<float4*>(&h0[(size_t)(row0 + rr) * HC_F + lane * 4]) = o;
    }
    __syncthreads();

    // --- GEMM2: h = sh(16x128) @ Wlin^T ; waves 0..3 each do 16 cols ---
    if (w < 4) {
        const int cb = w * 16;
        v8f acc = {};
        const float* wp = Wlin + (size_t)(cb + n) * HC_F;
        const float* ap = &sh[n * LDS_STRIDE];             // A rows from LDS (ds reads)
        for (int k = 0; k < HC_F; k += 4) {
            v2f a, b;
            a.x = ap[k + kb]; a.y = ap[k + kb + 1];
            b.x = wp[k + kb]; b.y = wp[k + kb + 1];
            acc = __builtin_amdgcn_wmma_f32_16x16x4_f32(false, a, false, b,
                                                        (short)0, acc, false, false);
        }
        #pragma unroll
        for (int r = 0; r < 8; ++r) {
            const int m = r + 8 * (lane >> 4);
            h[(size_t)(row0 + m) * HID_F + cb + n] = acc[r];
        }
    }
}

// ---------------------------------------------------------------------------
// Kernel 2: zero the agg accumulator
// ---------------------------------------------------------------------------
__global__ void k_zero(float4* __restrict__ p, int n4)
{
    int i = blockIdx.x * blockDim.x + threadIdx.x;
    if (i < n4) p[i] = make_float4(0.f, 0.f, 0.f, 0.f);
}

// ---------------------------------------------------------------------------
// Kernel 3: edge phase. One wave per edge. HID=64 -> 2 elems/lane.
// ---------------------------------------------------------------------------
__global__ __launch_bounds__(256) void k_edge(
    const int* __restrict__ ei, const float* __restrict__ h,
    const float* __restrict__ Watt, float* __restrict__ agg, int E)
{
    const int w    = threadIdx.x >> 5;
    const int lane = threadIdx.x & 31;
    const int e    = blockIdx.x * 8 + w;
    if (e >= E) return;

    const int r = ei[e];
    const int c = ei[E + e];
    const float* hr = h + (size_t)r * HID_F;
    const float* hc = h + (size_t)c * HID_F;
    const float hr0 = hr[lane], hr1 = hr[lane + 32];
    const float hc0 = hc[lane], hc1 = hc[lane + 32];

    const float bn0 = fmaxf(0.5f * hr0 + hc0, 0.f);
    const float bn1 = fmaxf(0.5f * hr1 + hc1, 0.f);
    float p0 = bn0 * Watt[lane]      + bn1 * Watt[32 + lane];
    float p1 = bn0 * Watt[64 + lane] + bn1 * Watt[96 + lane];
    wave_sum2(p0, p1);

    const float mx  = fmaxf(p0, p1);
    const float e0  = __expf(p0 - mx);
    const float e1  = __expf(p1 - mx);
    const float inv = 1.f / (e0 + e1);
    const float a0  = e0 * inv;
    const float a1  = e1 * inv;

    float* ap = agg + (size_t)r * HC_F;
    atomicAdd(ap + lane,       a0 * hc0);
    atomicAdd(ap + 32 + lane,  a0 * hc1);
    atomicAdd(ap + 64 + lane,  a1 * hc0);
    atomicAdd(ap + 96 + lane,  a1 * hc1);
}

// ---------------------------------------------------------------------------
// Kernel 4: tail. xh = 0.5*LN(relu(agg)) + 0.5*h0; out = xh @ W2^T + b2
// ---------------------------------------------------------------------------
__global__ __launch_bounds__(256) void k_tail(
    const float* __restrict__ agg, const float* __restrict__ h0,
    const float* __restrict__ g1, const float* __restrict__ beta1,
    const float* __restrict__ W2, const float* __restrict__ b2,
    float* __restrict__ out)
{
    __shared__ float sh[16 * LDS_STRIDE];
    const int w    = threadIdx.x >> 5;
    const int lane = threadIdx.x & 31;
    const int n    = lane & 15;
    const int kb   = (lane >> 4) * 2;
    const int row0 = blockIdx.x * 16;

    #pragma unroll
    for (int i = 0; i < 2; ++i) {
        const int rr = 2 * w + i;
        float4 v = *reinterpret_cast<const float4*>(&agg[(size_t)(row0 + rr) * HC_F + lane * 4]);
        v.x = fmaxf(v.x, 0.f); v.y = fmaxf(v.y, 0.f);
        v.z = fmaxf(v.z, 0.f); v.w = fmaxf(v.w, 0.f);
        float s = v.x + v.y + v.z + v.w;
        float q = v.x * v.x + v.y * v.y + v.z * v.z + v.w * v.w;
        wave_sum2(s, q);
        const float mu  = s * (1.f / 128.f);
        const float var = q * (1.f / 128.f) - mu * mu;
        const float rs  = rsqrtf(var + LN_EPS);
        const float4 g  = *reinterpret_cast<const float4*>(&g1[lane * 4]);
        const float4 bb = *reinterpret_cast<const float4*>(&beta1[lane * 4]);
        const float4 eg = *reinterpret_cast<const float4*>(&h0[(size_t)(row0 + rr) * HC_F + lane * 4]);
        float4 o;  // (1-BETA)*xh + BETA*ego with BETA=0.5
        o.x = 0.5f * ((v.x - mu) * rs * g.x + bb.x) + 0.5f * eg.x;
        o.y = 0.5f * ((v.y - mu) * rs * g.y + bb.y) + 0.5f * eg.y;
        o.z = 0.5f * ((v.z - mu) * rs * g.z + bb.z) + 0.5f * eg.z;
        o.w = 0.5f * ((v.w - mu) * rs * g.w + bb.w) + 0.5f * eg.w;
        *reinterpret_cast<float4*>(&sh[rr * LDS_STRIDE + lane * 4]) = o;
    }
    __syncthreads();

    // GEMM: 40 output cols -> 3 tiles of 16 (last 8 cols zero-masked). Waves 0..2.
    if (w < 3) {
        const int cb = w * 16;
        const int j  = cb + n;
        const float valid = (j < OUT_F) ? 1.f : 0.f;
        const int   j2    = (j < OUT_F) ? j : 0;
        const float* wp = W2 + (size_t)j2 * HC_F;
        const float* ap = &sh[n * LDS_STRIDE];
        v8f acc = {};
        for (int k = 0; k < HC_F; k += 4) {
            v2f a, b;
            a.x = ap[k + kb];         a.y = ap[k + kb + 1];
            b.x = wp[k + kb] * valid; b.y = wp[k + kb + 1] * valid;
            acc = __builtin_amdgcn_wmma_f32_16x16x4_f32(false, a, false, b,
                                                        (short)0, acc, false, false);
        }
        if (j < OUT_F) {
            const float bv = b2[j];
            #pragma unroll
            for (int r = 0; r < 8; ++r) {
                const int m = r + 8 * (lane >> 4);
                out[(size_t)(row0 + m) * OUT_F + j] = acc[r] + bv;
            }
        }
    }
}

// ---------------------------------------------------------------------------
extern "C" void kernel_launch(void* const* d_in, const int* in_sizes, int n_in,
                              void* d_out, int out_size, void* d_ws, size_t ws_size,
                              hipStream_t stream)
{
    const float* x     = (const float*)d_in[0];
    const int*   ei    = (const int*)  d_in[1];
    const float* W1    = (const float*)d_in[2];
    const float* b1    = (const float*)d_in[3];
    const float* g0    = (const float*)d_in[4];
    const float* beta0 = (const float*)d_in[5];
    const float* Wlin  = (const float*)d_in[6];
    const float* Watt  = (const float*)d_in[7];
    const float* g1    = (const float*)d_in[8];
    const float* beta1 = (const float*)d_in[9];
    const float* W2    = (const float*)d_in[10];
    const float* b2    = (const float*)d_in[11];
    float*       out   = (float*)d_out;

    const int N = in_sizes[0] / IN_F;      // 50000
    const int E = in_sizes[1] / 2;         // 800000

    // workspace carve: h0 [N,128] | h [N,64] | agg [N,128]
    float* h0  = (float*)d_ws;
    float* h   = h0 + (size_t)N * HC_F;
    float* agg = h  + (size_t)N * HID_F;

    const int nTiles = (N + 15) / 16;      // 3125

    // zero agg (independent of kernel 1, both precede edge phase)
    {
        const int n4 = (N * HC_F) / 4;
        k_zero<<<(n4 + 255) / 256, 256, 0, stream>>>((float4*)agg, n4);
    }
    k_gemm1_ln<<<nTiles, 256, 0, stream>>>(x, W1, b1, g0, beta0, Wlin, h0, h);
    k_edge<<<(E + 7) / 8, 256, 0, stream>>>(ei, h, Watt, agg, E);
    k_tail<<<nTiles, 256, 0, stream>>>(agg, h0, g1, beta1, W2, b2, out);
}